// nonLocalBlock_30966714204745
// MI455X (gfx1250) — compile-verified
//
#include <hip/hip_runtime.h>
#include <math.h>

typedef __attribute__((ext_vector_type(2))) float v2f;
typedef __attribute__((ext_vector_type(4))) float v4f;
typedef __attribute__((ext_vector_type(8))) float v8f;

#define BATCH 4
#define CIN   256
#define CHI   128
#define NPIX  4096   // 64*64

__device__ __forceinline__ v8f wmma4(v2f a, v2f b, v8f c) {
    // D = A(16x4 f32) * B(4x16 f32) + C(16x16 f32)
    return __builtin_amdgcn_wmma_f32_16x16x4_f32(
        /*neg_a=*/false, a, /*neg_b=*/false, b,
        /*c_mod=*/(short)0, c, /*reuse_a=*/false, /*reuse_b=*/false);
}

__device__ __forceinline__ v2f ld2(const float* p) { return *(const v2f*)p; }

// ---------------------------------------------------------------------------
// Kernel 1: fused theta/phi/gi 1x1-conv projections.
// out[mat][b][o][p] = sum_c W[o][c] * x[b][c][p] + bias[o]
// One wave -> 16(M=o) x 64(N=p) strip; K=CIN in steps of 4.
// waves = B(4) * 3 mats * 8 Mtiles * 64 strips = 6144 -> 768 blocks of 256.
// ---------------------------------------------------------------------------
__global__ void proj_kernel(const float* __restrict__ x,
                            const float* __restrict__ w_t, const float* __restrict__ b_t,
                            const float* __restrict__ w_f, const float* __restrict__ b_f,
                            const float* __restrict__ w_g, const float* __restrict__ b_g,
                            float* __restrict__ o_t, float* __restrict__ o_f,
                            float* __restrict__ o_g)
{
    int wid  = (blockIdx.x * blockDim.x + threadIdx.x) >> 5;
    int lane = threadIdx.x & 31;
    int half = lane >> 4, l16 = lane & 15;

    int nstrip = wid & 63;  wid >>= 6;
    int mt     = wid & 7;   wid >>= 3;
    int mat    = wid % 3;
    int b      = wid / 3;

    const float* W; const float* bias; float* out;
    if (mat == 0)      { W = w_t; bias = b_t; out = o_t; }
    else if (mat == 1) { W = w_f; bias = b_f; out = o_f; }
    else               { W = w_g; bias = b_g; out = o_g; }

    const float* X = x + (size_t)b * CIN * NPIX;
    out += (size_t)b * CHI * NPIX;

    int m0 = mt * 16;
    int n0 = nstrip * 64 + l16;

    v8f acc[4];
#pragma unroll
    for (int t = 0; t < 4; ++t)
#pragma unroll
        for (int v = 0; v < 8; ++v)
            acc[t][v] = bias[m0 + v + 8 * half];

    for (int k0 = 0; k0 < CIN; k0 += 4) {
        v2f a = ld2(&W[(m0 + l16) * CIN + k0 + 2 * half]);
#pragma unroll
        for (int t = 0; t < 4; ++t) {
            int n = n0 + 16 * t;
            v2f bb;
            bb.x = X[(k0 + 2 * half) * NPIX + n];
            bb.y = X[(k0 + 1 + 2 * half) * NPIX + n];
            acc[t] = wmma4(a, bb, acc[t]);
        }
    }
#pragma unroll
    for (int t = 0; t < 4; ++t) {
        int n = n0 + 16 * t;
#pragma unroll
        for (int v = 0; v < 8; ++v)
            out[(m0 + v + 8 * half) * NPIX + n] = acc[t][v];
    }
}

// ---------------------------------------------------------------------------
// Kernel 2: S[i][j] = sum_k theta_flat[i*CHI+k] * phi[k*NPIX+j]
// Workgroup-cooperative: block (8 waves) computes a 128(M) x 64(N) tile.
// The shared phi panel (128 K-rows x 64 cols, 32KB) is staged into LDS once
// per block with gfx1250 async global->LDS b128 copies (ASYNCcnt), then all
// waves feed their WMMA B fragments from LDS (ds_load).
// LDS row stride = 72 floats so the two half-wave K-rows (delta = 144 floats
// == 16 mod 64 banks) hit disjoint bank groups -> conflict-free ds reads.
// grid = 32 Mblocks * 64 Nblocks = 2048 blocks of 256.
// ---------------------------------------------------------------------------
#define BS_STRIDE 72

__global__ void gemm_scores(const float* __restrict__ theta,
                            const float* __restrict__ phi,
                            float* __restrict__ S)
{
    __shared__ float Bs[CHI * BS_STRIDE];   // 36 KB

    int tid  = threadIdx.x;
    int wave = tid >> 5;
    int lane = tid & 31;
    int half = lane >> 4, l16 = lane & 15;

    int nblk   = blockIdx.x & 63;           // 64 N-blocks of 64 cols
    int mblk   = blockIdx.x >> 6;           // 32 M-blocks of 128 rows
    int n_base = nblk * 64;
    int m0     = mblk * 128 + wave * 16;

    // ---- async-stage phi[0:128][n_base : n_base+64] into LDS ----
    {
        int r  = tid >> 4;                  // 16 rows per pass (16 threads/row)
        int c4 = (tid & 15) * 4;            // 16B chunk within the row
        const float* g = phi + (size_t)r * NPIX + n_base + c4;
        unsigned     l = (unsigned)(size_t)(&Bs[r * BS_STRIDE + c4]);
#pragma unroll
        for (int it = 0; it < 8; ++it) {    // 8 * 16 rows = 128 K-rows
            asm volatile("global_load_async_to_lds_b128 %0, %1, off"
                         :: "v"(l), "v"(g) : "memory");
            g += (size_t)16 * NPIX;
            l += 16 * BS_STRIDE * 4;
        }
        asm volatile("s_wait_asynccnt 0x0" ::: "memory");
    }
    __syncthreads();

    v8f zero = {0.f,0.f,0.f,0.f,0.f,0.f,0.f,0.f};
    v8f acc[4] = {zero, zero, zero, zero};

    for (int k0 = 0; k0 < CHI; k0 += 4) {
        v2f a = ld2(&theta[(m0 + l16) * CHI + k0 + 2 * half]);
        const float* brow0 = &Bs[(k0 + 2 * half) * BS_STRIDE];
        const float* brow1 = &Bs[(k0 + 1 + 2 * half) * BS_STRIDE];
#pragma unroll
        for (int t = 0; t < 4; ++t) {
            int n = l16 + 16 * t;
            v2f bb;
            bb.x = brow0[n];
            bb.y = brow1[n];
            acc[t] = wmma4(a, bb, acc[t]);
        }
    }
#pragma unroll
    for (int t = 0; t < 4; ++t) {
        int n = n_base + l16 + 16 * t;
#pragma unroll
        for (int v = 0; v < 8; ++v)
            S[(size_t)(m0 + v + 8 * half) * NPIX + n] = acc[t][v];
    }
}

// ---------------------------------------------------------------------------
// Softmax over axis=1 (columns of S). Online (flash-style) chunked stats.
// ---------------------------------------------------------------------------
__global__ void softmax_partial(const float* __restrict__ S,
                                float* __restrict__ pmax, float* __restrict__ psum)
{
    int j = blockIdx.x * blockDim.x + threadIdx.x;  // column, coalesced
    int c = blockIdx.y;                             // 64-row chunk id
    const float* p = S + (size_t)(c * 64) * NPIX + j;
    float mx = -INFINITY, sm = 0.f;
    for (int r = 0; r < 64; ++r) {
        float v  = p[(size_t)r * NPIX];
        float nm = fmaxf(mx, v);
        sm = sm * __expf(mx - nm) + __expf(v - nm);
        mx = nm;
    }
    pmax[c * NPIX + j] = mx;
    psum[c * NPIX + j] = sm;
}

__global__ void softmax_finalize(const float* __restrict__ pmax,
                                 const float* __restrict__ psum,
                                 float* __restrict__ cmax, float* __restrict__ cinv)
{
    int j = blockIdx.x * blockDim.x + threadIdx.x;
    float mx = -INFINITY;
    for (int c = 0; c < 64; ++c) mx = fmaxf(mx, pmax[c * NPIX + j]);
    float z = 0.f;
    for (int c = 0; c < 64; ++c) z += psum[c * NPIX + j] * __expf(pmax[c * NPIX + j] - mx);
    cmax[j] = mx;
    cinv[j] = 1.f / z;
}

__global__ void softmax_normalize(float* __restrict__ S,
                                  const float* __restrict__ cmax,
                                  const float* __restrict__ cinv)
{
    size_t idx = (size_t)blockIdx.x * blockDim.x + threadIdx.x;
    int j = (int)(idx & (NPIX - 1));
    S[idx] = __expf(S[idx] - cmax[j]) * cinv[j];
}

// ---------------------------------------------------------------------------
// Kernel 4: y_cn[c][i] = sum_j P[i][j] * gi_flat[j*CHI + c]
// One wave -> 16(M=i) x 32(N=c) strip (2 tiles); K=NPIX in steps of 4.
// D-tile stores are contiguous (two float4 per tile per lane) in (CH,N) layout.
// waves = 256 Mtiles * 4 strips = 1024 -> 128 blocks.
// ---------------------------------------------------------------------------
__global__ void gemm_av(const float* __restrict__ P,
                        const float* __restrict__ gi,
                        float* __restrict__ y)
{
    int wid  = (blockIdx.x * blockDim.x + threadIdx.x) >> 5;
    int lane = threadIdx.x & 31;
    int half = lane >> 4, l16 = lane & 15;

    int cs = wid & 3;               // c-strip (2 tiles of 16)
    int mt = wid >> 2;              // 0..255
    int m0 = mt * 16;
    int c0 = cs * 32 + l16;

    v8f zero = {0.f,0.f,0.f,0.f,0.f,0.f,0.f,0.f};
    v8f acc[2] = {zero, zero};

    for (int k0 = 0; k0 < NPIX; k0 += 4) {
        v2f a = ld2(&P[(size_t)(m0 + l16) * NPIX + k0 + 2 * half]);
#pragma unroll
        for (int t = 0; t < 2; ++t) {
            int c = c0 + 16 * t;
            v2f bb;
            bb.x = gi[(k0 + 2 * half) * CHI + c];
            bb.y = gi[(k0 + 1 + 2 * half) * CHI + c];
            acc[t] = wmma4(a, bb, acc[t]);
        }
    }
    int mbase = m0 + 8 * half;
#pragma unroll
    for (int t = 0; t < 2; ++t) {
        int c = c0 + 16 * t;
        v4f lo = {acc[t][0], acc[t][1], acc[t][2], acc[t][3]};
        v4f hi = {acc[t][4], acc[t][5], acc[t][6], acc[t][7]};
        *(v4f*)&y[(size_t)c * NPIX + mbase]     = lo;
        *(v4f*)&y[(size_t)c * NPIX + mbase + 4] = hi;
    }
}

// ---------------------------------------------------------------------------
// Kernel 5: out[o][p] = sum_k out_w[o][k] * y_cn[k][p] + out_b[o] + x[o][p]
// One wave -> 16(M=o) x 64(N=p) strip; K=CHI. waves = 16*64 = 1024 -> 128 blocks.
// ---------------------------------------------------------------------------
__global__ void gemm_out(const float* __restrict__ ow, const float* __restrict__ ob,
                         const float* __restrict__ y,  const float* __restrict__ xres,
                         float* __restrict__ out)
{
    int wid  = (blockIdx.x * blockDim.x + threadIdx.x) >> 5;
    int lane = threadIdx.x & 31;
    int half = lane >> 4, l16 = lane & 15;

    int nstrip = wid & 63;
    int mt     = wid >> 6;          // 0..15
    int m0 = mt * 16;
    int n0 = nstrip * 64 + l16;

    v8f acc[4];
#pragma unroll
    for (int t = 0; t < 4; ++t)
#pragma unroll
        for (int v = 0; v < 8; ++v)
            acc[t][v] = ob[m0 + v + 8 * half];

    for (int k0 = 0; k0 < CHI; k0 += 4) {
        v2f a = ld2(&ow[(m0 + l16) * CHI + k0 + 2 * half]);
#pragma unroll
        for (int t = 0; t < 4; ++t) {
            int n = n0 + 16 * t;
            v2f bb;
            bb.x = y[(k0 + 2 * half) * NPIX + n];
            bb.y = y[(k0 + 1 + 2 * half) * NPIX + n];
            acc[t] = wmma4(a, bb, acc[t]);
        }
    }
#pragma unroll
    for (int t = 0; t < 4; ++t) {
        int n = n0 + 16 * t;
#pragma unroll
        for (int v = 0; v < 8; ++v) {
            size_t idx = (size_t)(m0 + v + 8 * half) * NPIX + n;
            out[idx] = acc[t][v] + xres[idx];
        }
    }
}

// ---------------------------------------------------------------------------
extern "C" void kernel_launch(void* const* d_in, const int* in_sizes, int n_in,
                              void* d_out, int out_size, void* d_ws, size_t ws_size,
                              hipStream_t stream) {
    const float* x   = (const float*)d_in[0];
    const float* twt = (const float*)d_in[1];
    const float* tbb = (const float*)d_in[2];
    const float* fwt = (const float*)d_in[3];
    const float* fbb = (const float*)d_in[4];
    const float* gwt = (const float*)d_in[5];
    const float* gbb = (const float*)d_in[6];
    const float* owt = (const float*)d_in[7];
    const float* obb = (const float*)d_in[8];
    float* out = (float*)d_out;

    float* ws      = (float*)d_ws;
    float* t_theta = ws;                               // B*CHI*NPIX = 2097152
    float* t_phi   = t_theta + (size_t)BATCH * CHI * NPIX;
    float* t_gi    = t_phi   + (size_t)BATCH * CHI * NPIX;
    float* t_y     = t_gi    + (size_t)BATCH * CHI * NPIX;   // CHI*NPIX (per-batch reuse)
    float* t_S     = t_y     + (size_t)CHI * NPIX;           // NPIX*NPIX = 64MB (per-batch reuse, L2-resident)
    float* t_pm    = t_S     + (size_t)NPIX * NPIX;          // 64*NPIX
    float* t_ps    = t_pm    + (size_t)64 * NPIX;            // 64*NPIX
    float* t_cm    = t_ps    + (size_t)64 * NPIX;            // NPIX
    float* t_ci    = t_cm    + NPIX;                         // NPIX

    // projections for all batches / all three weight sets
    proj_kernel<<<768, 256, 0, stream>>>(x, twt, tbb, fwt, fbb, gwt, gbb,
                                         t_theta, t_phi, t_gi);

    for (int b = 0; b < BATCH; ++b) {
        const float* th = t_theta + (size_t)b * CHI * NPIX;
        const float* ph = t_phi   + (size_t)b * CHI * NPIX;
        const float* gg = t_gi    + (size_t)b * CHI * NPIX;

        gemm_scores<<<2048, 256, 0, stream>>>(th, ph, t_S);
        softmax_partial<<<dim3(NPIX / 256, 64), 256, 0, stream>>>(t_S, t_pm, t_ps);
        softmax_finalize<<<NPIX / 256, 256, 0, stream>>>(t_pm, t_ps, t_cm, t_ci);
        softmax_normalize<<<(NPIX * (NPIX / 256)), 256, 0, stream>>>(t_S, t_cm, t_ci);
        gemm_av<<<128, 256, 0, stream>>>(t_S, gg, t_y);
        gemm_out<<<128, 256, 0, stream>>>(owt, obb, t_y,
                                          x + (size_t)b * CIN * NPIX,
                                          out + (size_t)b * CIN * NPIX);
    }
}